// Conv_12970801234121
// MI455X (gfx1250) — compile-verified
//
#include <hip/hip_runtime.h>
#include <hip/hip_bf16.h>

#define V_N 100000
#define E_N 800000
#define B_N 8
#define FIN_N 64
#define FOUT_N 64
#define COLS 512            // B_N * FIN_N
#define M_ROWS (V_N * B_N)  // 800000 GEMM rows

typedef float v2f __attribute__((ext_vector_type(2)));
typedef float v8f __attribute__((ext_vector_type(8)));

// ---------------------------------------------------------------------------
// K1: x [B,FIN,V] (= [512, V] row-major, row c = b*FIN+f) -> x0 [V, 512]
// ---------------------------------------------------------------------------
__global__ void __launch_bounds__(256) k_transpose_in(const float* __restrict__ x,
                                                      float* __restrict__ x0) {
    __shared__ float t[32][33];
    const int v0 = blockIdx.x * 32;
    const int c0 = blockIdx.y * 32;
    for (int i = threadIdx.y; i < 32; i += 8)
        t[i][threadIdx.x] = x[(long)(c0 + i) * V_N + v0 + threadIdx.x];
    __syncthreads();
    for (int i = threadIdx.y; i < 32; i += 8)
        x0[(long)(v0 + i) * COLS + c0 + threadIdx.x] = t[threadIdx.x][i];
}

// ---------------------------------------------------------------------------
// K2: zero a buffer (float4), K4: d = -s (float4)
// ---------------------------------------------------------------------------
__global__ void __launch_bounds__(256) k_zero(float4* __restrict__ p, long n4) {
    long i = (long)blockIdx.x * blockDim.x + threadIdx.x;
    if (i < n4) { float4 z; z.x = 0.f; z.y = 0.f; z.z = 0.f; z.w = 0.f; p[i] = z; }
}
__global__ void __launch_bounds__(256) k_negcopy(const float4* __restrict__ s,
                                                 float4* __restrict__ d, long n4) {
    long i = (long)blockIdx.x * blockDim.x + threadIdx.x;
    if (i < n4) { float4 v = s[i]; v.x = -v.x; v.y = -v.y; v.z = -v.z; v.w = -v.w; d[i] = v; }
}

// ---------------------------------------------------------------------------
// K3/K5: COO scatter SpMM: dst[row,:] += scale * val * src[col,:]
// one wave (32 lanes) per edge; 512 columns -> 16 iterations
// ---------------------------------------------------------------------------
__global__ void __launch_bounds__(256) k_spmm_scatter(const float* __restrict__ src,
                                                      float* dst,
                                                      const int* __restrict__ rows,
                                                      const int* __restrict__ cols,
                                                      const float* __restrict__ vals,
                                                      float scale, int nE) {
    const int wave = threadIdx.x >> 5;
    const int lane = threadIdx.x & 31;
    const long e = (long)blockIdx.x * 8 + wave;
    if (e >= nE) return;
    const long row = rows[e];
    const long col = cols[e];
    const float a = vals[e] * scale;
    const float* s = src + col * COLS;
    float* d = dst + row * COLS;
#pragma unroll
    for (int i = 0; i < 16; ++i) {
        float v = s[i * 32 + lane];
        __hip_atomic_fetch_add(&d[i * 32 + lane], a * v,
                               __ATOMIC_RELAXED, __HIP_MEMORY_SCOPE_AGENT);
    }
}

// ---------------------------------------------------------------------------
// K6: GEMM  out[r, fo] = sum_k A_k[r, 0:64] * W_k[0:64, fo] + bias[fo]
//   A_k row r lives at xs_k[r*64 .. r*64+64)  (r = v*8+b)
//   out is written INTO the x0 buffer (flat-index-identical region; each wave
//   reads all of its A0 data into registers before any store).
//   One wave per 16-row x 64-col tile; V_WMMA_F32_16X16X4_F32, 48 steps/tile.
// Weight pre-swizzled in LDS: wl[k][kk][khalf][fo][pair], pair = two K-values
// consecutive so a B fragment is one ds_load_b64.
// ---------------------------------------------------------------------------
__global__ void __launch_bounds__(256) k_gemm_wmma(const float* x0, /* aliased w/ out */
                                                   const float* __restrict__ x1,
                                                   const float* __restrict__ x2,
                                                   const float* __restrict__ w,
                                                   const float* __restrict__ bias,
                                                   float* out) {
    __shared__ __align__(16) float wl[3 * 16 * 2 * 64 * 2];  // 12288 floats = 48KB
    for (int t = threadIdx.x; t < 3 * 64 * 64; t += 256) {
        int k = t >> 12;          // /4096
        int rem = t & 4095;
        int f = rem >> 6;
        int fo = rem & 63;
        int kk = f >> 2;
        int khalf = (f >> 1) & 1;
        int pair = f & 1;
        wl[((((k * 16 + kk) * 2 + khalf) * 64) + fo) * 2 + pair] = w[t];
    }
    __syncthreads();

    const int wave = threadIdx.x >> 5;
    const int lane = threadIdx.x & 31;
    const long tile = (long)blockIdx.x * 8 + wave;  // 50000 tiles total
    const int n = lane & 15;      // A row index / B,D column index
    const int khalf = lane >> 4;  // which K pair of the 4-deep step

    const float* pk0 = x0 + tile * 1024;
    const float* pk1 = x1 + tile * 1024;
    const float* pk2 = x2 + tile * 1024;

    v8f acc[4];
#pragma unroll
    for (int j = 0; j < 4; ++j) acc[j] = (v8f){0.f, 0.f, 0.f, 0.f, 0.f, 0.f, 0.f, 0.f};

#pragma unroll
    for (int k = 0; k < 3; ++k) {
        const float* A = (k == 0) ? pk0 : (k == 1) ? pk1 : pk2;
#pragma unroll
        for (int kk = 0; kk < 16; ++kk) {
            // A fragment: lane(<16) m=n holds K = kk*4 + {0,1}; lanes 16..31 K = kk*4 + {2,3}
            v2f a = *(const v2f*)(A + n * 64 + kk * 4 + khalf * 2);
            const float* wb = &wl[(((k * 16 + kk) * 2 + khalf) * 64) * 2];
#pragma unroll
            for (int j = 0; j < 4; ++j) {
                v2f b = *(const v2f*)(wb + (j * 16 + n) * 2);
                acc[j] = __builtin_amdgcn_wmma_f32_16x16x4_f32(
                    false, a, false, b, (short)0, acc[j], false, false);
            }
        }
    }

    // D layout: VGPR i -> row i (lanes 0-15) / row 8+i (lanes 16-31), col = lane&15
    float* po = out + tile * 1024;
#pragma unroll
    for (int j = 0; j < 4; ++j) {
        float bs = bias[j * 16 + n];
#pragma unroll
        for (int i = 0; i < 8; ++i) {
            int row = khalf ? (8 + i) : i;
            po[row * 64 + j * 16 + n] = acc[j][i] + bs;
        }
    }
}

// ---------------------------------------------------------------------------
// K7: out_tmp[(v*8+b)*64 + fo]  ->  d_out[b*64*V + fo*V + v]   (LDS tiled)
// ---------------------------------------------------------------------------
__global__ void __launch_bounds__(256) k_transpose_out(const float* __restrict__ ot,
                                                       float* __restrict__ out) {
    __shared__ float t[32][33];
    const int v0 = blockIdx.x * 32;
    const int fo0 = blockIdx.y * 32;
    const int b = blockIdx.z;
    for (int i = threadIdx.y; i < 32; i += 8)
        t[i][threadIdx.x] = ot[((long)(v0 + i) * 8 + b) * 64 + fo0 + threadIdx.x];
    __syncthreads();
    for (int i = threadIdx.y; i < 32; i += 8)
        out[(long)b * 64 * V_N + (long)(fo0 + i) * V_N + v0 + threadIdx.x] = t[threadIdx.x][i];
}

// ---------------------------------------------------------------------------
extern "C" void kernel_launch(void* const* d_in, const int* in_sizes, int n_in,
                              void* d_out, int out_size, void* d_ws, size_t ws_size,
                              hipStream_t stream) {
    const float* x    = (const float*)d_in[0];   // [B, FIN, V]
    const int*   rows = (const int*)d_in[1];     // [E]
    const int*   cols = (const int*)d_in[2];     // [E]
    const float* vals = (const float*)d_in[3];   // [E]
    const float* w    = (const float*)d_in[4];   // [K, FIN, FOUT]
    const float* bias = (const float*)d_in[5];   // [FOUT]
    float* out = (float*)d_out;                  // [B, FOUT, V]

    const size_t bufElems = (size_t)V_N * COLS;  // 51.2M floats each
    float* x0 = (float*)d_ws;                    // also reused as GEMM output
    float* x1 = x0 + bufElems;
    float* x2 = x1 + bufElems;
    (void)ws_size; (void)n_in; (void)in_sizes; (void)out_size; // needs 3*bufElems*4 bytes

    const long n4 = (long)bufElems / 4;
    const int zb = (int)((n4 + 255) / 256);

    // x0 = permuted x
    k_transpose_in<<<dim3(V_N / 32, COLS / 32), dim3(32, 8), 0, stream>>>(x, x0);
    // x1 = L @ x0
    k_zero<<<zb, 256, 0, stream>>>((float4*)x1, n4);
    k_spmm_scatter<<<(E_N + 7) / 8, 256, 0, stream>>>(x0, x1, rows, cols, vals, 1.0f, E_N);
    // x2 = 2 L @ x1 - x0
    k_negcopy<<<zb, 256, 0, stream>>>((const float4*)x0, (float4*)x2, n4);
    k_spmm_scatter<<<(E_N + 7) / 8, 256, 0, stream>>>(x1, x2, rows, cols, vals, 2.0f, E_N);
    // GEMM (+bias) -> writes into x0 buffer (safe aliasing, see kernel comment)
    k_gemm_wmma<<<(M_ROWS / 16) / 8, 256, 0, stream>>>(x0, x1, x2, w, bias, x0);
    // permute to [B, FOUT, V]
    k_transpose_out<<<dim3(V_N / 32, FOUT_N / 32, B_N), dim3(32, 8), 0, stream>>>(x0, out);
}